// SpikeFP32RMSNormFullFP64_84713934946541
// MI455X (gfx1250) — compile-verified
//
#include <hip/hip_runtime.h>

// SpikeFP32RMSNormFullFP64 for MI455X (gfx1250).
// Bandwidth-bound: 512 MiB total HBM traffic -> ~22 us floor @ 23.3 TB/s.
// Strategy: async DMA each 16 KB row HBM->LDS (CDNA5 global_load_async_to_lds,
// tracked by ASYNCcnt), do fp64 reduction + fp64 scaling out of LDS (x is read
// from HBM exactly once), stream fp32 results back with non-temporal stores so
// the output doesn't evict the input stream from the 192 MB L2.

typedef __attribute__((ext_vector_type(4))) float v4f;

namespace {
constexpr int kDim = 4096;           // feature dimension
constexpr int kTPB = 256;            // 8 wave32s per workgroup
constexpr int kVecPT = kDim / (kTPB * 4);  // 4 float4 chunks per thread
constexpr int kWaves = kTPB / 32;
constexpr double kEps = 1e-6;
}  // namespace

__global__ __launch_bounds__(kTPB) void rmsnorm_fp64acc_async(
    const float* __restrict__ x, const float* __restrict__ w,
    float* __restrict__ y) {
  __shared__ float tile[kDim];          // 16 KB staged row (LDS, 320 KB/WGP)
  __shared__ double wave_part[kWaves];  // per-wave partial sums
  __shared__ double inv_rms_s;          // broadcast slot

  const size_t row = blockIdx.x;
  const float* __restrict__ xrow = x + row * (size_t)kDim;
  float* __restrict__ yrow = y + row * (size_t)kDim;
  const int t = threadIdx.x;

  // ---- Phase 0: async DMA HBM -> LDS, 128b per lane, bypasses VGPRs. ----
  // Generic pointers to __shared__ carry the LDS byte offset in their low 32
  // bits (ISA: LDS aperture mapping discards upper bits), so we can hand the
  // VDST lds-offset VGPR and the 64-bit global VADDR pair straight to the
  // CDNA5 async-load instruction.
#pragma unroll
  for (int i = 0; i < kVecPT; ++i) {
    const int v4 = t + i * kTPB;  // float4 index within the row
    unsigned lds_off = (unsigned)(size_t)&tile[v4 * 4];
    unsigned long long gaddr = (unsigned long long)(size_t)(xrow + v4 * 4);
    asm volatile("global_load_async_to_lds_b128 %0, %1, off"
                 :
                 : "v"(lds_off), "v"(gaddr)
                 : "memory");
  }
  asm volatile("s_wait_asynccnt 0x0" ::: "memory");  // our copies landed in LDS
  __syncthreads();  // publish all waves' tiles to the workgroup

  // ---- Phase 1: fp64 sum of squares (decode fp32 -> fp64, square, sum). ----
  const v4f* tile4 = reinterpret_cast<const v4f*>(tile);
  double ss = 0.0;
#pragma unroll
  for (int i = 0; i < kVecPT; ++i) {
    v4f v = tile4[t + i * kTPB];  // ds_load_b128
    double a = (double)v.x, b = (double)v.y, c = (double)v.z, d = (double)v.w;
    ss = fma(a, a, ss);
    ss = fma(b, b, ss);
    ss = fma(c, c, ss);
    ss = fma(d, d, ss);
  }
  // wave32 butterfly reduction (warpSize == 32 on gfx1250)
#pragma unroll
  for (int o = 16; o > 0; o >>= 1) ss += __shfl_xor(ss, o, 32);
  if ((t & 31) == 0) wave_part[t >> 5] = ss;
  __syncthreads();
  if (t == 0) {
    double tot = 0.0;
#pragma unroll
    for (int i = 0; i < kWaves; ++i) tot += wave_part[i];
    inv_rms_s = 1.0 / sqrt(tot / (double)kDim + kEps);  // exact fp64 path
  }
  __syncthreads();
  const double inv = inv_rms_s;

  // ---- Phase 2: fp64 scale * weight, encode fp32, stream out (NT). ----
  const v4f* w4 = reinterpret_cast<const v4f*>(w);  // 16 KB, L2-resident
  v4f* y4 = reinterpret_cast<v4f*>(yrow);
#pragma unroll
  for (int i = 0; i < kVecPT; ++i) {
    const int idx = t + i * kTPB;
    v4f v = tile4[idx];
    v4f g = w4[idx];
    v4f o;
    o.x = (float)((double)v.x * inv * (double)g.x);
    o.y = (float)((double)v.y * inv * (double)g.y);
    o.z = (float)((double)v.z * inv * (double)g.z);
    o.w = (float)((double)v.w * inv * (double)g.w);
    __builtin_nontemporal_store(o, &y4[idx]);  // global_store_b128 th:NT
  }
}

extern "C" void kernel_launch(void* const* d_in, const int* in_sizes, int n_in,
                              void* d_out, int out_size, void* d_ws,
                              size_t ws_size, hipStream_t stream) {
  const float* x = (const float*)d_in[0];
  const float* w = (const float*)d_in[1];
  float* y = (float*)d_out;
  const int rows = in_sizes[0] / kDim;  // 4 * 4096 = 16384 rows
  rmsnorm_fp64acc_async<<<rows, kTPB, 0, stream>>>(x, w, y);
}